// IFVD_82351702934116
// MI455X (gfx1250) — compile-verified
//
#include <hip/hip_runtime.h>
#include <math.h>

// ---------------- problem constants (from reference) ----------------
#define B_    8
#define C_    512
#define H_    64
#define W_    64
#define HS_   32
#define WS_   32
#define NCLS  150
#define NPAD  160          // classes padded to multiple of 16 for WMMA tiles
#define EPS_MEAN 1e-6f
#define EPS_COS  1e-8f

typedef float v2f __attribute__((ext_vector_type(2)));
typedef float v8f __attribute__((ext_vector_type(8)));

__device__ __forceinline__ int iclamp(int v, int lo, int hi) {
    return v < lo ? lo : (v > hi ? hi : v);
}

// bilinear (half-pixel, edge-clamped) source setup for 32 -> 64 upsample
__device__ __forceinline__ void bl_setup(int o, int& i0, int& i1, float& f) {
    float src = 0.5f * (float)o - 0.25f;
    float fl  = floorf(src);
    f  = src - fl;
    i0 = iclamp((int)fl,     0, 31);
    i1 = iclamp((int)fl + 1, 0, 31);
}

// ---------------- k1: per-(batch,class) pixel counts ----------------
__global__ void k1_counts(const int* __restrict__ target, float* __restrict__ cnt) {
    int b = blockIdx.x;
    __shared__ int hist[NPAD];
    for (int i = threadIdx.x; i < NPAD; i += 256) hist[i] = 0;
    __syncthreads();
    for (int p = threadIdx.x; p < H_ * W_; p += 256) {
        int cls = target[b * H_ * W_ + p];
        atomicAdd(&hist[cls], 1);
    }
    __syncthreads();
    for (int i = threadIdx.x; i < NPAD; i += 256)
        cnt[b * NPAD + i] = (float)hist[i];
}

// ---------------- k2: per-(b, 4-channel chunk) class sums for T and bilinear-S ----------------
// grid = B * (C/4) = 1024 blocks, 256 threads; writes EVERY entry of sums (incl. padded classes)
__global__ void k2_sums(const float* __restrict__ pS, const float* __restrict__ pT,
                        const int* __restrict__ target,
                        float* __restrict__ sumsT, float* __restrict__ sumsS) {
    int b  = blockIdx.x >> 7;            // / (C/4)
    int c0 = (blockIdx.x & 127) * 4;
    int t  = threadIdx.x;
    int cl = t >> 6;                     // 0..3 local channel
    int w  = t & 63;
    int c  = c0 + cl;

    __shared__ float binsT[4][NPAD];
    __shared__ float binsS[4][NPAD];
    for (int i = t; i < 4 * NPAD; i += 256) {
        (&binsT[0][0])[i] = 0.f;
        (&binsS[0][0])[i] = 0.f;
    }
    __syncthreads();

    int w0, w1; float fw; bl_setup(w, w0, w1, fw);

    const float* Tb = pT + ((size_t)(b * C_ + c)) * (H_ * W_);
    const float* Sb = pS + ((size_t)(b * C_ + c)) * (HS_ * WS_);
    const int*   tg = target + b * H_ * W_;

    for (int h = 0; h < H_; ++h) {
        int cls  = tg[h * W_ + w];
        float vT = Tb[h * W_ + w];
        atomicAdd(&binsT[cl][cls], vT);

        int h0, h1; float fh; bl_setup(h, h0, h1, fh);
        float a00 = Sb[h0 * WS_ + w0], a01 = Sb[h0 * WS_ + w1];
        float a10 = Sb[h1 * WS_ + w0], a11 = Sb[h1 * WS_ + w1];
        float vS = (1.f - fh) * ((1.f - fw) * a00 + fw * a01)
                 +        fh  * ((1.f - fw) * a10 + fw * a11);
        atomicAdd(&binsS[cl][cls], vS);
    }
    __syncthreads();

    for (int i = t; i < 4 * NPAD; i += 256) {
        int cli = i / NPAD, cls = i % NPAD;
        size_t o = ((size_t)(b * NPAD + cls)) * C_ + (c0 + cli);
        sumsT[o] = binsT[cli][cls];
        sumsS[o] = binsS[cli][cls];
    }
}

// ---------------- k3: ||sums[b,cls]|| via WMMA f32 Gram diagonal ----------------
// grid = 2 * B * (NPAD/16) = 160 blocks of 1 wave (32 threads).
// For a Gram matrix the documented f32 16x16x4 A and B lane layouts coincide:
//   lane l: row/col = l&15 ; VGPR pair holds K = k0 + 2*(l>>4) + {0,1}
// so a single b64 load per lane feeds both the A and B operands.
__global__ void k3_center_norms(const float* __restrict__ sumsT, const float* __restrict__ sumsS,
                                float* __restrict__ snT, float* __restrict__ snS) {
    int id   = blockIdx.x;
    int selS = (id >= B_ * (NPAD / 16));
    int bt   = selS ? (id - B_ * (NPAD / 16)) : id;
    int b    = bt / (NPAD / 16);
    int tile = bt % (NPAD / 16);

    const float* sums = selS ? sumsS : sumsT;
    float*       out  = selS ? snS   : snT;

    int l    = threadIdx.x;           // 0..31, full wave active (EXEC all-1 for WMMA)
    int row  = l & 15;
    int koff = (l >> 4) * 2;

    const float* rp = sums + ((size_t)(b * NPAD + tile * 16 + row)) * C_ + koff;

    v8f acc = {};
    for (int k0 = 0; k0 < C_; k0 += 4) {
        v2f a = *(const v2f*)(rp + k0);   // 8B-aligned: offset is even
        acc = __builtin_amdgcn_wmma_f32_16x16x4_f32(
            /*neg_a=*/false, a, /*neg_b=*/false, a,
            /*c_mod=*/(short)0, acc, /*reuse_a=*/false, /*reuse_b=*/false);
    }

    // extract diagonal: D(i,i) lives in lane (i<8 ? i : i+16), vgpr (i&7)
    __shared__ float g[32 * 8];
#pragma unroll
    for (int r = 0; r < 8; ++r) g[l * 8 + r] = acc[r];
    __syncthreads();
    if (l < 16) {
        int srcLane = (l < 8) ? l : (l + 16);
        float d = g[srcLane * 8 + (l & 7)];
        out[b * NPAD + tile * 16 + l] = sqrtf(d);
    }
}

// ---------------- k4: per-pixel cosine for S and T, per-block partial loss ----------------
// grid = B*H = 512 blocks, 256 threads: thread = (channel-group cg 0..3, pixel w 0..63)
__global__ void k4_cos(const float* __restrict__ pS, const float* __restrict__ pT,
                       const int* __restrict__ target,
                       const float* __restrict__ sumsT, const float* __restrict__ sumsS,
                       const float* __restrict__ cnt,
                       const float* __restrict__ snT, const float* __restrict__ snS,
                       float* __restrict__ partial) {
    int blk = blockIdx.x;
    int b = blk >> 6, h = blk & 63;
    int t = threadIdx.x;
    int w = t & 63, cg = t >> 6;

    int cls = target[b * H_ * W_ + h * W_ + w];
    const float* sT = sumsT + ((size_t)(b * NPAD + cls)) * C_;
    const float* sS = sumsS + ((size_t)(b * NPAD + cls)) * C_;

    int w0, w1; float fw; bl_setup(w, w0, w1, fw);
    int h0, h1; float fh; bl_setup(h, h0, h1, fh);
    float w00 = (1.f - fh) * (1.f - fw), w01 = (1.f - fh) * fw;
    float w10 = fh * (1.f - fw),         w11 = fh * fw;

    size_t tOff  = (((size_t)(b * C_)) * H_ + h) * W_ + w;   // + c * (H_*W_)
    size_t sBase = ((size_t)(b * C_)) * (HS_ * WS_);          // + c * (HS_*WS_)

    float dotT = 0.f, nfT = 0.f, dotS = 0.f, nfS = 0.f;
    for (int c = cg; c < C_; c += 4) {
        const float* tp = pT + tOff + (size_t)c * (H_ * W_);
        __builtin_prefetch(tp + 4 * (H_ * W_), 0, 1);         // -> global_prefetch_b8
        float vT = *tp;
        dotT = fmaf(vT, sT[c], dotT);
        nfT  = fmaf(vT, vT, nfT);

        const float* Sb = pS + sBase + (size_t)c * (HS_ * WS_);
        float vS = w00 * Sb[h0 * WS_ + w0] + w01 * Sb[h0 * WS_ + w1]
                 + w10 * Sb[h1 * WS_ + w0] + w11 * Sb[h1 * WS_ + w1];
        dotS = fmaf(vS, sS[c], dotS);
        nfS  = fmaf(vS, vS, nfS);
    }

    __shared__ float4 red[256];
    red[t] = make_float4(dotT, nfT, dotS, nfS);
    __syncthreads();

    float acc = 0.f;
    if (cg == 0) {
        float4 r0 = red[w], r1 = red[w + 64], r2 = red[w + 128], r3 = red[w + 192];
        float DT = r0.x + r1.x + r2.x + r3.x;
        float NT = r0.y + r1.y + r2.y + r3.y;
        float DS = r0.z + r1.z + r2.z + r3.z;
        float NS = r0.w + r1.w + r2.w + r3.w;
        // center = sums/d  =>  cos = dotRaw / max(||f||*||sums||, eps*d)
        float d  = cnt[b * NPAD + cls] + EPS_MEAN;
        float cT = DT / fmaxf(sqrtf(NT) * snT[b * NPAD + cls], EPS_COS * d);
        float cS = DS / fmaxf(sqrtf(NS) * snS[b * NPAD + cls], EPS_COS * d);
        float diff = cS - cT;
        acc = diff * diff;
    }
    __syncthreads();

    __shared__ float rsum[256];
    rsum[t] = acc;
    __syncthreads();
    for (int s = 128; s > 0; s >>= 1) {
        if (t < s) rsum[t] += rsum[t + s];
        __syncthreads();
    }
    if (t == 0) partial[blk] = rsum[0];
}

// ---------------- k5: deterministic final reduction ----------------
__global__ void k5_final(const float* __restrict__ partial, float* __restrict__ out) {
    __shared__ float s[256];
    int t = threadIdx.x;
    s[t] = partial[t] + partial[t + 256];
    __syncthreads();
    for (int k = 128; k > 0; k >>= 1) {
        if (t < k) s[t] += s[t + k];
        __syncthreads();
    }
    if (t == 0) out[0] = s[0] * (100.0f / (float)(B_ * H_ * W_));
}

// ---------------- launch ----------------
extern "C" void kernel_launch(void* const* d_in, const int* in_sizes, int n_in,
                              void* d_out, int out_size, void* d_ws, size_t ws_size,
                              hipStream_t stream) {
    const float* preds_S = (const float*)d_in[0];   // [8,512,32,32]
    const float* preds_T = (const float*)d_in[1];   // [8,512,64,64]
    const int*   target  = (const int*)d_in[2];     // [8,1,64,64]
    float* out = (float*)d_out;

    // workspace layout (floats)
    float* ws     = (float*)d_ws;
    float* sumsT  = ws;                                     // B*NPAD*C = 655360
    float* sumsS  = sumsT + (size_t)B_ * NPAD * C_;         // 655360
    float* cnt    = sumsS + (size_t)B_ * NPAD * C_;         // 1280
    float* snT    = cnt   + (size_t)B_ * NPAD;              // 1280
    float* snS    = snT   + (size_t)B_ * NPAD;              // 1280
    float* part   = snS   + (size_t)B_ * NPAD;              // 512
    (void)in_sizes; (void)n_in; (void)out_size; (void)ws_size;

    k1_counts<<<B_, 256, 0, stream>>>(target, cnt);
    k2_sums<<<B_ * (C_ / 4), 256, 0, stream>>>(preds_S, preds_T, target, sumsT, sumsS);
    k3_center_norms<<<2 * B_ * (NPAD / 16), 32, 0, stream>>>(sumsT, sumsS, snT, snS);
    k4_cos<<<B_ * H_, 256, 0, stream>>>(preds_S, preds_T, target,
                                        sumsT, sumsS, cnt, snT, snS, part);
    k5_final<<<1, 256, 0, stream>>>(part, out);
}